// GPCANet_5334349382410
// MI455X (gfx1250) — compile-verified
//
#include <hip/hip_runtime.h>

// GPCANet for MI455X (gfx1250).
// - Working set (~90MB) is L2-resident (192MB L2); SpMM scatter-adds are the
//   dominant cost -> float4 gathers + global_atomic_add_f32, 32-bit index math.
// - Dense GEMMs: fp32 WMMA (v_wmma_f32_16x16x4_f32), W slice transposed into
//   LDS (bank-padded) so B fragments are single ds_load_b64 per k-step.

namespace {

constexpr int kN = 50000;
constexpr int kE = 800000;
constexpr int kFeat = 128;
constexpr int kHid = 64;
constexpr int kClass = 40;
constexpr int kTrain = 5000;
// ALPHA=1.0, BETA=0.1 -> a1=a2=0.5
constexpr float kC1 = 0.45f;   // a1*(1-beta)  multiplies SpMM result
constexpr float kC2 = 0.50f;   // a2           multiplies centered x
constexpr float kCsc = 0.05f;  // a1*beta      multiplies label-projection term

typedef float v2f __attribute__((ext_vector_type(2)));
typedef float v8f __attribute__((ext_vector_type(8)));

__global__ void zero_f32(float* __restrict__ p, int n) {
  int i = blockIdx.x * blockDim.x + threadIdx.x;
  if (i < n) p[i] = 0.0f;
}

__global__ void zero_f32x4(float4* __restrict__ p, int n4) {
  int i = blockIdx.x * blockDim.x + threadIdx.x;
  if (i < n4) p[i] = make_float4(0.f, 0.f, 0.f, 0.f);
}

// Column sums: blockDim.x == D (64 or 128); grid-strided over rows; coalesced.
__global__ void colsum(const float* __restrict__ x, float* __restrict__ s, int n) {
  int t = threadIdx.x;
  int d = blockDim.x;
  float acc = 0.0f;
  for (int i = blockIdx.x; i < n; i += gridDim.x) acc += x[(size_t)i * d + t];
  atomicAdd(&s[t], acc);
}

// xc = x - colsum/N ; z = xc  (z0 of the power iteration)
template <int D>
__global__ void center(const float* __restrict__ x, const float* __restrict__ csum,
                       float* __restrict__ xc, float* __restrict__ z, int total) {
  int i = blockIdx.x * blockDim.x + threadIdx.x;
  if (i >= total) return;
  float v = x[i] - csum[i & (D - 1)] * (1.0f / (float)kN);
  xc[i] = v;
  z[i] = v;
}

__global__ void class_counts(const int* __restrict__ labels, const int* __restrict__ tidx,
                             float* __restrict__ counts) {
  int e = blockIdx.x * blockDim.x + threadIdx.x;
  if (e < kTrain) atomicAdd(&counts[labels[tidx[e]]], 1.0f);
}

__global__ void norm_accum(const int* __restrict__ labels, const int* __restrict__ tidx,
                           const float* __restrict__ counts, float* __restrict__ norm) {
  int e = blockIdx.x * blockDim.x + threadIdx.x;
  if (e < kTrain) {
    int i = tidx[e];
    atomicAdd(&norm[i], counts[labels[i]]);
  }
}

__global__ void norm_invert(float* __restrict__ norm, int n) {
  int i = blockIdx.x * blockDim.x + threadIdx.x;
  if (i < n) norm[i] = 1.0f / (norm[i] + 1e-8f);
}

// part1: acc[row] += w * z[col]   (D/4 threads per edge, float4 gather, 4 atomics)
template <int D>
__global__ void spmm_atomic(const int* __restrict__ erow, const int* __restrict__ ecol,
                            const float* __restrict__ ew, const float* __restrict__ z,
                            float* __restrict__ acc, int nedges) {
  constexpr int CH = D / 4;
  int g = blockIdx.x * blockDim.x + threadIdx.x;
  int e = g / CH;
  if (e >= nedges) return;
  int p = (g % CH) * 4;
  int r = erow[e];
  int c = ecol[e];
  float w = ew[e];
  const float4 zv = *(const float4*)(z + c * D + p);  // offsets fit in 32-bit
  float* dst = acc + r * D + p;
  atomicAdd(dst + 0, w * zv.x);
  atomicAdd(dst + 1, w * zv.y);
  atomicAdd(dst + 2, w * zv.z);
  atomicAdd(dst + 3, w * zv.w);
}

// Ytz[c] += z[train_idx[t]]  where c = labels[train_idx[t]]
template <int D>
__global__ void ytz_accum(const int* __restrict__ labels, const int* __restrict__ tidx,
                          const float* __restrict__ z, float* __restrict__ ytz) {
  constexpr int CH = D / 4;
  int g = blockIdx.x * blockDim.x + threadIdx.x;
  int t = g / CH;
  if (t >= kTrain) return;
  int p = (g % CH) * 4;
  int i = tidx[t];
  int c = labels[i];
  const float4 zv = *(const float4*)(z + i * D + p);
  float* dst = ytz + c * D + p;
  atomicAdd(dst + 0, zv.x);
  atomicAdd(dst + 1, zv.y);
  atomicAdd(dst + 2, zv.z);
  atomicAdd(dst + 3, zv.w);
}

// znew = a1*(1-beta)*part1 + a2*xc   (float4)
__global__ void combine4(const float4* __restrict__ xc, float4* __restrict__ znew, int n4) {
  int i = blockIdx.x * blockDim.x + threadIdx.x;
  if (i >= n4) return;
  float4 a = znew[i];
  float4 b = xc[i];
  a.x = kC1 * a.x + kC2 * b.x;
  a.y = kC1 * a.y + kC2 * b.y;
  a.z = kC1 * a.z + kC2 * b.z;
  a.w = kC1 * a.w + kC2 * b.w;
  znew[i] = a;
}

// znew[train_idx[t]] += a1*beta * Ytz[labels[train_idx[t]]] * rnorm[train_idx[t]]
template <int D>
__global__ void scatter_part2(const int* __restrict__ labels, const int* __restrict__ tidx,
                              const float* __restrict__ ytz, const float* __restrict__ rnorm,
                              float* __restrict__ znew) {
  constexpr int CH = D / 4;
  int g = blockIdx.x * blockDim.x + threadIdx.x;
  int t = g / CH;
  if (t >= kTrain) return;
  int p = (g % CH) * 4;
  int i = tidx[t];
  int c = labels[i];
  float s = rnorm[i] * kCsc;
  const float4 yv = *(const float4*)(ytz + c * D + p);
  float* dst = znew + i * D + p;
  atomicAdd(dst + 0, yv.x * s);
  atomicAdd(dst + 1, yv.y * s);
  atomicAdd(dst + 2, yv.z * s);
  atomicAdd(dst + 3, yv.w * s);
}

// out[N,H] = Z[N,D] @ W[D,H] + bias (+ReLU), fp32 WMMA 16x16x4.
// Block = 256 threads = 8 waves; each wave owns one 16x16 output tile
// (8 consecutive M-tiles per block, one H-tile per blockIdx.y).
// The block's 16-column slice of W is transposed into LDS (stride D+4 to
// spread banks): B fragment = one aligned ds_load_b64 per k-step.
// A frag (16x4, MxK): lane -> m = lane&15, K pair kb = 2*(lane>>4).
// C/D (16x16): VGPR i of lane -> D[i + 8*(lane>>4)][lane&15].
template <int D, int H, int RELU>
__global__ void wmma_gemm_lds(const float* __restrict__ Z, const float* __restrict__ W,
                              const float* __restrict__ bias, float* __restrict__ out,
                              int mtiles) {
  constexpr int LSTRIDE = D + 4;
  __shared__ float wt[16 * LSTRIDE];

  const int tid = threadIdx.x;
  const int gcol0 = blockIdx.y * 16;

  // Cooperative fill: wt[c16][k] = W[k][gcol0 + c16], zero-padded past H.
  for (int i = tid; i < 16 * D; i += 256) {
    int c16 = i / D;  // compile-time D -> shift
    int k = i % D;
    int gcol = gcol0 + c16;
    wt[c16 * LSTRIDE + k] = (gcol < H) ? W[k * H + gcol] : 0.0f;
  }
  __syncthreads();

  const int wave = tid >> 5;
  const int lane = tid & 31;
  const int tileM = blockIdx.x * 8 + wave;
  if (tileM >= mtiles) return;

  const int m = lane & 15;
  const int kb = (lane >> 4) << 1;
  const float* zrow = Z + (size_t)(tileM * 16 + m) * D;
  const float* brow = wt + m * LSTRIDE + kb;  // m == lane&15 == B column index

  v8f acc = {};
#pragma unroll
  for (int k0 = 0; k0 < D; k0 += 4) {
    v2f a = *(const v2f*)(zrow + k0 + kb);  // 8B aligned (k0%4==0, kb in {0,2})
    v2f b = *(const v2f*)(brow + k0);       // ds_load_b64, conflict-free
    acc = __builtin_amdgcn_wmma_f32_16x16x4_f32(false, a, false, b, (short)0, acc,
                                                false, false);
  }

  const int col = gcol0 + m;
  if (col < H) {
    float bv = bias[col];
    int mbase = tileM * 16 + ((lane >> 4) << 3);
#pragma unroll
    for (int i = 0; i < 8; ++i) {
      float v = acc[i] + bv;
      if (RELU) v = fmaxf(v, 0.0f);
      out[(size_t)(mbase + i) * H + col] = v;
    }
  }
}

template <int D, int H, int RELU>
void run_layer(const float* xin, const float* W, const float* bias, float* outbuf,
               float* z0, float* z1, float* xc, float* mean, float* ytz,
               const float* rnorm,
               const int* erow, const int* ecol, const float* ew,
               const int* labels, const int* tidx, hipStream_t stream) {
  constexpr int kTotal = kN * D;
  constexpr int kEdgeThreads = kE * (D / 4);
  constexpr int kTrainThreads = kTrain * (D / 4);
  constexpr int kMTiles = kN / 16;

  zero_f32<<<1, 128, 0, stream>>>(mean, D);
  colsum<<<256, D, 0, stream>>>(xin, mean, kN);
  center<D><<<(kTotal + 255) / 256, 256, 0, stream>>>(xin, mean, xc, z0, kTotal);

  float* z = z0;
  float* znew = z1;
  for (int it = 0; it < 10; ++it) {
    zero_f32x4<<<(kTotal / 4 + 255) / 256, 256, 0, stream>>>((float4*)znew, kTotal / 4);
    zero_f32<<<(kClass * D + 255) / 256, 256, 0, stream>>>(ytz, kClass * D);
    spmm_atomic<D><<<(kEdgeThreads + 255) / 256, 256, 0, stream>>>(erow, ecol, ew, z,
                                                                   znew, kE);
    ytz_accum<D><<<(kTrainThreads + 255) / 256, 256, 0, stream>>>(labels, tidx, z, ytz);
    combine4<<<(kTotal / 4 + 255) / 256, 256, 0, stream>>>((const float4*)xc,
                                                           (float4*)znew, kTotal / 4);
    scatter_part2<D><<<(kTrainThreads + 255) / 256, 256, 0, stream>>>(labels, tidx, ytz,
                                                                      rnorm, znew);
    float* tmp = z; z = znew; znew = tmp;
  }

  dim3 g((kMTiles + 7) / 8, (H + 15) / 16);
  wmma_gemm_lds<D, H, RELU><<<g, 256, 0, stream>>>(z, W, bias, outbuf, kMTiles);
}

}  // namespace

extern "C" void kernel_launch(void* const* d_in, const int* in_sizes, int n_in,
                              void* d_out, int out_size, void* d_ws, size_t ws_size,
                              hipStream_t stream) {
  (void)in_sizes; (void)n_in; (void)out_size; (void)ws_size;

  const float* x     = (const float*)d_in[0];
  const float* ew    = (const float*)d_in[1];
  const float* W1    = (const float*)d_in[2];
  const float* b1    = (const float*)d_in[3];
  const float* W2    = (const float*)d_in[4];
  const float* b2    = (const float*)d_in[5];
  const float* Wout  = (const float*)d_in[6];
  const float* bout  = (const float*)d_in[7];
  const int* erow    = (const int*)d_in[8];
  const int* ecol    = (const int*)d_in[9];
  const int* labels  = (const int*)d_in[10];
  const int* tidx    = (const int*)d_in[11];
  float* out = (float*)d_out;

  // Workspace layout (floats). Total ~22.5M floats (~90 MB) -> fits 192MB L2.
  float* ws     = (float*)d_ws;
  float* z0     = ws;                                // N*128
  float* z1     = z0 + (size_t)kN * kFeat;           // N*128
  float* xc     = z1 + (size_t)kN * kFeat;           // N*128
  float* h      = xc + (size_t)kN * kFeat;           // N*64
  float* mean   = h + (size_t)kN * kHid;             // 128
  float* counts = mean + kFeat;                      // 64 (>= NCLASS)
  float* rnorm  = counts + 64;                       // N
  float* ytz    = rnorm + kN;                        // NCLASS*128

  // Label statistics (same for both layers): counts[c], rnorm[i] = 1/(norm+1e-8)
  zero_f32<<<1, 64, 0, stream>>>(counts, kClass);
  zero_f32<<<(kN + 255) / 256, 256, 0, stream>>>(rnorm, kN);
  class_counts<<<(kTrain + 255) / 256, 256, 0, stream>>>(labels, tidx, counts);
  norm_accum<<<(kTrain + 255) / 256, 256, 0, stream>>>(labels, tidx, counts, rnorm);
  norm_invert<<<(kN + 255) / 256, 256, 0, stream>>>(rnorm, kN);

  // Layer 1: x[N,128] -> h[N,64] (ReLU)
  run_layer<kFeat, kHid, 1>(x, W1, b1, h, z0, z1, xc, mean, ytz, rnorm,
                            erow, ecol, ew, labels, tidx, stream);
  // Layer 2: h[N,64] -> h[N,64] (ReLU); h is consumed by centering before GEMM writes it
  run_layer<kHid, kHid, 1>(h, W2, b2, h, z0, z1, xc, mean, ytz, rnorm,
                           erow, ecol, ew, labels, tidx, stream);

  // Head: out = h @ Wout + bout  (no ReLU), H = 40 -> last 16-col tile masked
  constexpr int kMTiles = kN / 16;
  dim3 g((kMTiles + 7) / 8, (kClass + 15) / 16);
  wmma_gemm_lds<kHid, kClass, 0><<<g, 256, 0, stream>>>(h, Wout, bout, out, kMTiles);
}